// _GraphBasicBlock_79937931313499
// MI455X (gfx1250) — compile-verified
//
#include <hip/hip_runtime.h>

typedef __attribute__((ext_vector_type(2))) float v2f;
typedef __attribute__((ext_vector_type(8))) float v8f;

#define DIN 128
// LDS layout for B operands: k-pair interleaved, padded pair stride.
//   s_w[(k>>1)*PSTR + n*2 + (k&1)]  ->  B fragment {k0,k0+1} at column n is one
//   contiguous 8B ds_load_b64. PSTR=288 floats => pair stride 1152B == 128 mod
//   256B, so lanes 0-15 (banks 0-31) and lanes 16-31 (banks 32-63) never clash.
#define PSTR 288
#define NPAIR (DIN / 2)

// ---------------------------------------------------------------------------
// Zero the workspace (agg sums + degree counts).
// ---------------------------------------------------------------------------
__global__ void zero_ws_kernel(float* __restrict__ ws, long long n) {
  long long i = (long long)blockIdx.x * blockDim.x + threadIdx.x;
  long long stride = (long long)gridDim.x * blockDim.x;
  for (; i < n; i += stride) ws[i] = 0.0f;
}

// ---------------------------------------------------------------------------
// Edge scatter: one wave32 per edge. Lane l loads float4 of x[src] (coalesced
// 512B per edge) and atomically adds into agg[dst]; lane 0 counts degree.
// __hip_atomic_fetch_add (relaxed, agent scope, result unused) lowers to
// no-return global_atomic_add_f32; agg (51.2MB) lives in the 192MB L2.
// ---------------------------------------------------------------------------
__global__ void __launch_bounds__(256)
scatter_kernel(const float* __restrict__ x,
               const long long* __restrict__ ei,
               float* __restrict__ agg,
               float* __restrict__ deg,
               int E) {
  int warp = (int)((blockIdx.x * blockDim.x + threadIdx.x) >> 5);
  int lane = threadIdx.x & 31;
  if (warp >= E) return;
  long long src = ei[warp];
  long long dst = ei[(long long)E + warp];
  float4 v = ((const float4*)(x + src * DIN))[lane];
  float* ar = agg + dst * DIN + lane * 4;
  (void)__hip_atomic_fetch_add(ar + 0, v.x, __ATOMIC_RELAXED, __HIP_MEMORY_SCOPE_AGENT);
  (void)__hip_atomic_fetch_add(ar + 1, v.y, __ATOMIC_RELAXED, __HIP_MEMORY_SCOPE_AGENT);
  (void)__hip_atomic_fetch_add(ar + 2, v.z, __ATOMIC_RELAXED, __HIP_MEMORY_SCOPE_AGENT);
  (void)__hip_atomic_fetch_add(ar + 3, v.w, __ATOMIC_RELAXED, __HIP_MEMORY_SCOPE_AGENT);
  if (lane == 0)
    (void)__hip_atomic_fetch_add(deg + dst, 1.0f, __ATOMIC_RELAXED, __HIP_MEMORY_SCOPE_AGENT);
}

// ---------------------------------------------------------------------------
// Fused GEMM: out = (agg/deg) @ W_l^T + b_l + x @ W_r^T
// 8 waves/block; each wave owns a 16-node row block and all 8 column tiles.
// WMMA fp32 16x16x4; both matrix paths and the bias fused into one
// accumulator set (8 x v8f = 64 VGPRs).
//
// Fragment layouts (ISA 7.12.2, 32-bit):
//   A:  lane l, vgpr v : m = l&15, k = 4s + v + 2*(l>>4)  (contiguous float2)
//   B:  lane l, vgpr v : n = l&15, k = 4s + v + 2*(l>>4)  (one ds_load_b64)
//   C/D:lane l, vgpr v : n = l&15, m = v + 8*(l>>4)
// ---------------------------------------------------------------------------
__global__ void __launch_bounds__(256)
sage_gemm_kernel(const float* __restrict__ agg,
                 const float* __restrict__ deg,
                 const float* __restrict__ x,
                 const float* __restrict__ Wl,
                 const float* __restrict__ bl,
                 const float* __restrict__ Wr,
                 float* __restrict__ out,
                 int N) {
  __shared__ float s_wl[NPAIR * PSTR];
  __shared__ float s_wr[NPAIR * PSTR];

  int tid = threadIdx.x;
  // Cooperative staging: coalesced row-major reads of W[j][k]; scatter into the
  // pair-interleaved LDS layout (write conflicts are a one-time 128KB cost).
  for (int i = tid; i < DIN * DIN; i += 256) {
    int j = i >> 7;     // output row of W
    int k = i & 127;    // input col of W
    int a = (k >> 1) * PSTR + j * 2 + (k & 1);
    s_wl[a] = Wl[i];
    s_wr[a] = Wr[i];
  }
  __syncthreads();

  int wave = tid >> 5;
  int lane = tid & 31;
  int row_base = (blockIdx.x * 8 + wave) * 16;
  if (row_base >= N) return;  // wave-uniform: EXEC stays all-1s for WMMA

  int lo = lane & 15;   // A-row m within tile / output col n within tile
  int hi = lane >> 4;   // k-half selector

  // Mean normalization folded into the A-fragment load.
  float rscale = 1.0f / fmaxf(deg[row_base + lo], 1.0f);

  v8f acc[8];
#pragma unroll
  for (int t = 0; t < 8; ++t) {
    float b = bl[t * 16 + lo];  // n = t*16+lo is constant across the 8 vgprs
    acc[t] = (v8f){b, b, b, b, b, b, b, b};
  }

  const float* aggrow = agg + (long long)(row_base + lo) * DIN;
  const float* xrow   = x   + (long long)(row_base + lo) * DIN;

#pragma unroll 4
  for (int s = 0; s < 32; ++s) {
    int k0 = 4 * s + 2 * hi;                    // even -> 8B aligned everywhere
    v2f a_agg = *(const v2f*)(aggrow + k0);
    a_agg = a_agg * rscale;
    v2f a_x = *(const v2f*)(xrow + k0);
    const v2f* bL = (const v2f*)(s_wl + (k0 >> 1) * PSTR + lo * 2);
    const v2f* bR = (const v2f*)(s_wr + (k0 >> 1) * PSTR + lo * 2);
#pragma unroll
    for (int t = 0; t < 8; ++t) {
      acc[t] = __builtin_amdgcn_wmma_f32_16x16x4_f32(
          false, a_agg, false, bL[t * 16], (short)0, acc[t], false, false);
      acc[t] = __builtin_amdgcn_wmma_f32_16x16x4_f32(
          false, a_x, false, bR[t * 16], (short)0, acc[t], false, false);
    }
  }

  // Store: lanes 0-15 write row m=v, lanes 16-31 row m=v+8 (64B segments).
#pragma unroll
  for (int t = 0; t < 8; ++t) {
    int n = t * 16 + lo;
#pragma unroll
    for (int v = 0; v < 8; ++v) {
      int m = v + 8 * hi;
      out[(long long)(row_base + m) * DIN + n] = acc[t][v];
    }
  }
}

// ---------------------------------------------------------------------------
// Launch
// ---------------------------------------------------------------------------
extern "C" void kernel_launch(void* const* d_in, const int* in_sizes, int n_in,
                              void* d_out, int out_size, void* d_ws, size_t ws_size,
                              hipStream_t stream) {
  const float*     x  = (const float*)d_in[0];
  const long long* ei = (const long long*)d_in[1];  // int64 edge_index [2, E]
  const float*     Wl = (const float*)d_in[2];
  const float*     bl = (const float*)d_in[3];
  const float*     Wr = (const float*)d_in[4];
  float* out = (float*)d_out;

  int N = in_sizes[0] / DIN;
  int E = in_sizes[1] / 2;

  float* agg = (float*)d_ws;                   // N * 128 floats
  float* deg = agg + (long long)N * DIN;       // N floats

  long long zn = (long long)N * DIN + N;
  zero_ws_kernel<<<2048, 256, 0, stream>>>(agg, zn);

  int edges_per_block = 256 / 32;
  int sblocks = (E + edges_per_block - 1) / edges_per_block;
  scatter_kernel<<<sblocks, 256, 0, stream>>>(x, ei, agg, deg, E);

  int row_tiles = (N + 15) / 16;
  int gblocks = (row_tiles + 7) / 8;
  sage_gemm_kernel<<<gblocks, 256, 0, stream>>>(agg, deg, x, Wl, bl, Wr, out, N);
}